// LabelPropagationLoss_62045097558058
// MI455X (gfx1250) — compile-verified
//
#include <hip/hip_runtime.h>

#define NN 8192
#define NSTEPS 10
#define ALPHA 0.5f
#define EPSV 1e-10f

typedef __attribute__((ext_vector_type(2))) float v2f;
typedef __attribute__((ext_vector_type(8))) float v8f;

// ---------------------------------------------------------------------------
// Kernel 1 (bulk of traffic): fused  out = M .* adj  AND  d[row] = rsqrt(rowsum(adj))
// One block per row, float4-vectorized streaming. 768 MB total -> HBM bound.
// ---------------------------------------------------------------------------
__global__ __launch_bounds__(256) void fused_mul_degree_kernel(
    const float* __restrict__ adj, const float* __restrict__ M,
    float* __restrict__ out, float* __restrict__ d) {
  const int row = blockIdx.x;
  const long base = (long)row * NN;
  const int tid = threadIdx.x;

  const float4* a4 = (const float4*)(adj + base);
  const float4* m4 = (const float4*)(M + base);
  float4* o4 = (float4*)(out + base);

  float sum = 0.0f;
#pragma unroll
  for (int i = 0; i < NN / (4 * 256); ++i) {   // 8 iters of float4
    const int idx = tid + i * 256;
    float4 a = a4[idx];
    float4 m = m4[idx];
    sum += (a.x + a.y) + (a.z + a.w);
    float4 o;
    o.x = a.x * m.x; o.y = a.y * m.y; o.z = a.z * m.z; o.w = a.w * m.w;
    o4[idx] = o;
  }

  __shared__ float red[256];
  red[tid] = sum;
  __syncthreads();
#pragma unroll
  for (int s = 128; s > 0; s >>= 1) {
    if (tid < s) red[tid] += red[tid + s];
    __syncthreads();
  }
  if (tid == 0) d[row] = rsqrtf(red[0]);   // sum^-0.5
}

// ---------------------------------------------------------------------------
// Kernel 2: v = 1 - pos ; w = d .* v ; zero the loss accumulators
// ---------------------------------------------------------------------------
__global__ void init_kernel(const float* __restrict__ pos,
                            const float* __restrict__ d,
                            float* __restrict__ v, float* __restrict__ w,
                            float* __restrict__ accum) {
  const int i = blockIdx.x * blockDim.x + threadIdx.x;
  if (i < NN) {
    float vv = 1.0f - pos[i];
    v[i] = vv;
    w[i] = d[i] * vv;
  }
  if (i == 0) { accum[0] = 0.0f; accum[1] = 0.0f; }
}

// ---------------------------------------------------------------------------
// Kernel 3 (x10): one propagation step via V_WMMA_F32_16X16X4_F32.
//   u = A @ w ;  v' = ALPHA * d .* u + (1-ALPHA) * v ;  w' = d .* v'
// Block = 16 rows, 8 waves; wave `wid` reduces K-slice [wid*1024, wid*1024+1024).
// A-operand layout (f32 16x4): lane<16 holds A[m][k0..k0+1], lane>=16 holds
// A[m][k0+2..k0+3] -> a single b64 load per lane. B is w replicated across the
// 16 N columns: identical b64 load pattern from w. D column 0 = partial GEMV.
// ---------------------------------------------------------------------------
__global__ __launch_bounds__(256) void step_kernel(
    const float* __restrict__ A, const float* __restrict__ w_in,
    const float* __restrict__ v_in, const float* __restrict__ d,
    float* __restrict__ v_out, float* __restrict__ w_out) {
  const int row_base = blockIdx.x * 16;
  const int tid = threadIdx.x;
  const int wid = tid >> 5;        // wave in block: 0..7
  const int lane = tid & 31;
  const int m = lane & 15;         // row within tile
  const int khalf = lane >> 4;     // 0: K=k0,k0+1   1: K=k0+2,k0+3

  const int kslice = wid * (NN / 8);             // 1024-wide K slice per wave
  const float* __restrict__ arow = A + (long)(row_base + m) * NN;

  v8f c = {};
#pragma unroll 8
  for (int kb = 0; kb < NN / 8; kb += 4) {
    const int k0 = kslice + kb + 2 * khalf;
    v2f a = *(const v2f*)(arow + k0);            // global_load_b64 (A tile)
    v2f b = *(const v2f*)(w_in + k0);            // global_load_b64 (vector, L2-hot)
    // D = A(16x4) x B(4x16) + C ; every N-column of D equals the partial GEMV
    c = __builtin_amdgcn_wmma_f32_16x16x4_f32(
        /*neg_a=*/false, a, /*neg_b=*/false, b,
        /*c_mod=*/(short)0, c, /*reuse_a=*/false, /*reuse_b=*/false);
  }

  // Extract column N=0: lane 0 holds M=0..7 in c[0..7], lane 16 holds M=8..15.
  __shared__ float part[8][16];
  if (lane == 0) {
#pragma unroll
    for (int r = 0; r < 8; ++r) part[wid][r] = c[r];
  } else if (lane == 16) {
#pragma unroll
    for (int r = 0; r < 8; ++r) part[wid][8 + r] = c[r];
  }
  __syncthreads();

  if (tid < 16) {
    float u = 0.0f;
#pragma unroll
    for (int q = 0; q < 8; ++q) u += part[q][tid];   // cross-wave K reduction
    const int row = row_base + tid;
    const float dv = d[row];
    const float vn = ALPHA * dv * u + (1.0f - ALPHA) * v_in[row];
    v_out[row] = vn;
    w_out[row] = dv * vn;        // pre-scale for next step's GEMV
  }
}

// ---------------------------------------------------------------------------
// Kernel 4: loss reduction  accum[0] += sum(pos*log(1-v+eps)), accum[1] += sum(pos)
// ---------------------------------------------------------------------------
__global__ __launch_bounds__(256) void loss_kernel(const float* __restrict__ pos,
                                                   const float* __restrict__ v,
                                                   float* __restrict__ accum) {
  const int tid = threadIdx.x;
  const int i = blockIdx.x * blockDim.x + tid;
  float num = 0.0f, den = 0.0f;
  if (i < NN) {
    const float p = pos[i];
    num = p * logf(1.0f - v[i] + EPSV);
    den = p;
  }
  __shared__ float r0[256];
  __shared__ float r1[256];
  r0[tid] = num; r1[tid] = den;
  __syncthreads();
#pragma unroll
  for (int s = 128; s > 0; s >>= 1) {
    if (tid < s) { r0[tid] += r0[tid + s]; r1[tid] += r1[tid + s]; }
    __syncthreads();
  }
  if (tid == 0) {
    atomicAdd(&accum[0], r0[0]);
    atomicAdd(&accum[1], r1[0]);
  }
}

__global__ void finalize_kernel(const float* __restrict__ accum,
                                float* __restrict__ out_loss) {
  out_loss[0] = -accum[0] / accum[1];
}

// ---------------------------------------------------------------------------
extern "C" void kernel_launch(void* const* d_in, const int* in_sizes, int n_in,
                              void* d_out, int out_size, void* d_ws, size_t ws_size,
                              hipStream_t stream) {
  const float* adj = (const float*)d_in[0];
  const float* pos = (const float*)d_in[1];
  const float* M   = (const float*)d_in[2];

  float* out_adj  = (float*)d_out;                 // N*N floats
  float* out_loss = out_adj + (size_t)NN * NN;     // 1 float

  // Workspace layout: d | v0 | w0 | v1 | w1 | accum[2]   (~160 KB)
  float* ws = (float*)d_ws;
  float* d_deg = ws;
  float* v0 = ws + 1 * NN;
  float* w0 = ws + 2 * NN;
  float* v1 = ws + 3 * NN;
  float* w1 = ws + 4 * NN;
  float* accum = ws + 5 * NN;

  // 1) updated_adj + degrees in one streaming pass over adj/M
  fused_mul_degree_kernel<<<NN, 256, 0, stream>>>(adj, M, out_adj, d_deg);

  // 2) v0 = 1 - pos ; w0 = d .* v0 ; zero accumulators
  init_kernel<<<NN / 256, 256, 0, stream>>>(pos, d_deg, v0, w0, accum);

  // 3) 10 propagation steps (WMMA GEMV), ping-pong v/w
  float* vin = v0; float* win = w0; float* vout = v1; float* wout = w1;
  for (int s = 0; s < NSTEPS; ++s) {
    step_kernel<<<NN / 16, 256, 0, stream>>>(adj, win, vin, d_deg, vout, wout);
    float* t;
    t = vin; vin = vout; vout = t;
    t = win; win = wout; wout = t;
  }

  // 4) loss = -(sum pos*log(1-v+eps)) / sum(pos)
  loss_kernel<<<NN / 256, 256, 0, stream>>>(pos, vin, accum);
  finalize_kernel<<<1, 1, 0, stream>>>(accum, out_loss);
}